// GraphIsomorphismNetwork_70970039599401
// MI455X (gfx1250) — compile-verified
//
#include <hip/hip_runtime.h>

// CDNA5 / gfx1250. wave32.
typedef float v2f __attribute__((ext_vector_type(2)));
typedef float v8f __attribute__((ext_vector_type(8)));

#define BN_EPS 1e-5f

// -------- agg = x with zero-padded row stride Kp (Kp = roundup4(F)) ----------------------
// grid.x = N (one block per row), block = 256
__global__ __launch_bounds__(256)
void gin_copy_pad_kernel(const float* __restrict__ x, float* __restrict__ agg,
                         int F, int Kp) {
    const int row = blockIdx.x;
    const float* __restrict__ xr = x + (size_t)row * F;
    float* __restrict__ ar = agg + (size_t)row * Kp;
    for (int c = threadIdx.x; c < Kp; c += 256)
        ar[c] = (c < F) ? xr[c] : 0.0f;
}

// -------- scatter-add over edges: one wave per edge, lanes stripe features ---------------
__global__ __launch_bounds__(256)
void gin_edge_kernel(const float* __restrict__ x, const long long* __restrict__ edges,
                     float* __restrict__ agg, int E, int F, int Kp) {
    int e = blockIdx.x * 8 + (threadIdx.x >> 5);   // 8 wave32s per block
    if (e >= E) return;
    int lane = threadIdx.x & 31;
    long long src = edges[e];                 // edges[0][e]
    long long dst = edges[(long long)E + e];  // edges[1][e]
    const float* __restrict__ xr = x + (long long)src * F;
    float* __restrict__ ar = agg + (long long)dst * Kp;
    for (int f = lane; f < F; f += 32)
        atomicAdd(&ar[f], xr[f]);             // global_atomic_add_f32, L2-resident
}

// -------- repack W[K,F] -> Wt quad layout: Wt[((k>>2)*F + c)*4 + (k&3)], zero-pad K->Kp --
__global__ void gin_packW_kernel(const float* __restrict__ W, float* __restrict__ Wt,
                                 int K, int Kp, int F) {
    int i = blockIdx.x * blockDim.x + threadIdx.x;   // over Kp*F
    if (i >= Kp * F) return;
    int k = i / F, c = i - k * F;
    float v = (k < K) ? W[(size_t)k * F + c] : 0.0f;
    Wt[(((size_t)(k >> 2) * F + c) << 2) + (k & 3)] = v;
}

// -------- C[N,F] = A[N,Kp] @ W + bias via V_WMMA_F32_16X16X4_F32 -------------------------
// Block = 128 threads = 4 waves; wave owns rows [rowBase,rowBase+16), block owns 64 cols.
// Each wave: 1 A fragment feeds 4 col-tiles (4 accumulators).
// Fragment layouts per ISA 7.12.2:
//   A 16x4: lane%16 = row, lane/16 picks K pair; {a.x,a.y} = K {lo,hi}  (one b64 load)
//   B 4x16: lane%16 = col, same K-pair indexing (Wt quad layout -> one b64 load)
//   C/D:    VGPR r -> row r + 8*(lane/16), lane%16 -> col
__global__ __launch_bounds__(128)
void gin_gemm_wmma_kernel(const float* __restrict__ A, const float* __restrict__ Wt,
                          const float* __restrict__ bias, float* __restrict__ C,
                          int N, int Kp, int F) {
    const int lane  = threadIdx.x & 31;
    const int wave  = threadIdx.x >> 5;
    const int lmod  = lane & 15;
    const int lhalf = lane >> 4;
    const int colBase = blockIdx.x * 64;
    const int rowBase = blockIdx.y * 64 + wave * 16;

    int arow = rowBase + lmod;
    if (arow >= N) arow = N - 1;              // clamp: compute a valid duplicate row, drop at store

    const float* __restrict__ pa = A + (size_t)arow * Kp + 2 * lhalf;          // 8B aligned
    const float* __restrict__ pb = Wt + (((size_t)(colBase + lmod)) << 2) + 2 * lhalf;

    v8f acc[4] = {v8f{}, v8f{}, v8f{}, v8f{}};
    const int quads = Kp >> 2;
#pragma unroll 4
    for (int q = 0; q < quads; ++q) {
        v2f a  = *(const v2f*)pa;
        v2f b0 = *(const v2f*)(pb);
        v2f b1 = *(const v2f*)(pb + 64);      // +16 cols * 4
        v2f b2 = *(const v2f*)(pb + 128);
        v2f b3 = *(const v2f*)(pb + 192);
        acc[0] = __builtin_amdgcn_wmma_f32_16x16x4_f32(false, a, false, b0, (short)0, acc[0], false, false);
        acc[1] = __builtin_amdgcn_wmma_f32_16x16x4_f32(false, a, false, b1, (short)0, acc[1], false, false);
        acc[2] = __builtin_amdgcn_wmma_f32_16x16x4_f32(false, a, false, b2, (short)0, acc[2], false, false);
        acc[3] = __builtin_amdgcn_wmma_f32_16x16x4_f32(false, a, false, b3, (short)0, acc[3], false, false);
        pa += 4;
        pb += (size_t)F << 2;                 // next K-quad block
    }

#pragma unroll
    for (int t = 0; t < 4; ++t) {
        const int col = colBase + 16 * t + lmod;
        const float bv = bias[col];
#pragma unroll
        for (int r = 0; r < 8; ++r) {
            int rr = rowBase + r + 8 * lhalf;
            if (rr < N) C[(size_t)rr * F + col] = acc[t][r] + bv;
        }
    }
}

// -------- per-column sum / sum-of-squares, deterministic two-stage -----------------------
__global__ __launch_bounds__(256)
void gin_colstats_kernel(const float* __restrict__ h, int N, int F,
                         float* __restrict__ partials) {
    const int col = blockIdx.x * 64 + (threadIdx.x & 63);
    const int rt  = threadIdx.x >> 6;                  // 0..3
    float s = 0.0f, s2 = 0.0f;
    for (int r = blockIdx.y * 4 + rt; r < N; r += 4 * gridDim.y) {
        float v = h[(size_t)r * F + col];
        s += v; s2 += v * v;
    }
    __shared__ float ls[256], ls2[256];
    ls[threadIdx.x] = s; ls2[threadIdx.x] = s2;
    __syncthreads();
    if (rt == 0) {
        int t = threadIdx.x;
        float ts  = ls[t]  + ls[t + 64]  + ls[t + 128]  + ls[t + 192];
        float ts2 = ls2[t] + ls2[t + 64] + ls2[t + 128] + ls2[t + 192];
        size_t base = (size_t)blockIdx.y * 2 * F;
        partials[base + col]     = ts;
        partials[base + F + col] = ts2;
    }
}

__global__ void gin_reduce_kernel(const float* __restrict__ partials, int npart, int twoF,
                                  float* __restrict__ sums) {
    int i = blockIdx.x * blockDim.x + threadIdx.x;
    if (i >= twoF) return;
    float s = 0.0f;
    for (int p = 0; p < npart; ++p) s += partials[(size_t)p * twoF + i];
    sums[i] = s;
}

// -------- in-place BN (batch stats, biased var like jnp.var) + ReLU ----------------------
__global__ void gin_bnrelu_kernel(float* __restrict__ h, const float* __restrict__ sums,
                                  const float* __restrict__ gamma, const float* __restrict__ beta,
                                  int N, int Fmask, int F, int total) {
    int i = blockIdx.x * blockDim.x + threadIdx.x;
    if (i >= total) return;
    int col = i & Fmask;                       // F is a power of 2 (128/256/512)
    float invN = 1.0f / (float)N;
    float mu  = sums[col] * invN;
    float var = sums[F + col] * invN - mu * mu;
    float y = (h[i] - mu) * (1.0f / sqrtf(var + BN_EPS)) * gamma[col] + beta[col];
    h[i] = fmaxf(y, 0.0f);
}

extern "C" void kernel_launch(void* const* d_in, const int* in_sizes, int n_in,
                              void* d_out, int out_size, void* d_ws, size_t ws_size,
                              hipStream_t stream) {
    const float*     v     = (const float*)d_in[0];
    const long long* edges = (const long long*)d_in[1];   // jnp.int64, shape (2, E)
    const float* W1 = (const float*)d_in[2],  *b1 = (const float*)d_in[3];
    const float* g1 = (const float*)d_in[4],  *be1 = (const float*)d_in[5];
    const float* W2 = (const float*)d_in[6],  *b2 = (const float*)d_in[7];
    const float* g2 = (const float*)d_in[8],  *be2 = (const float*)d_in[9];
    const float* W3 = (const float*)d_in[10], *b3 = (const float*)d_in[11];
    const float* g3 = (const float*)d_in[12], *be3 = (const float*)d_in[13];

    const int N = in_sizes[0] / 86;   // 100000
    const int E = in_sizes[1] / 2;    // 3200000
    const int NPART = 128;

    // workspace: agg [N*256] | Wt [256*512] | partials [NPART*2*512] | sums [2*512]
    float* agg      = (float*)d_ws;
    float* Wt       = agg + (size_t)N * 256;
    float* partials = Wt + (size_t)256 * 512;
    float* sums     = partials + (size_t)NPART * 1024;

    float* x1 = (float*)d_out;                 // [N,128]
    float* x2 = x1 + (size_t)N * 128;          // [N,256]
    float* x3 = x2 + (size_t)N * 256;          // [N,512]

    auto layer = [&](const float* xin, int Fin, int Fout,
                     const float* W, const float* b,
                     const float* g, const float* be, float* xout) {
        const int Kp = (Fin + 3) & ~3;         // 88, 128, 256
        gin_copy_pad_kernel<<<N, 256, 0, stream>>>(xin, agg, Fin, Kp);
        gin_edge_kernel<<<(E + 7) / 8, 256, 0, stream>>>(xin, edges, agg, E, Fin, Kp);
        gin_packW_kernel<<<(Kp * Fout + 255) / 256, 256, 0, stream>>>(W, Wt, Fin, Kp, Fout);
        dim3 gg(Fout / 64, (N + 63) / 64);
        gin_gemm_wmma_kernel<<<gg, 128, 0, stream>>>(agg, Wt, b, xout, N, Kp, Fout);
        dim3 gs(Fout / 64, NPART);
        gin_colstats_kernel<<<gs, 256, 0, stream>>>(xout, N, Fout, partials);
        gin_reduce_kernel<<<(2 * Fout + 255) / 256, 256, 0, stream>>>(
            partials, NPART, 2 * Fout, sums);
        const int tot = N * Fout;
        gin_bnrelu_kernel<<<(tot + 255) / 256, 256, 0, stream>>>(
            xout, sums, g, be, N, Fout - 1, Fout, tot);
    };

    layer(v,  86,  128, W1, b1, g1, be1, x1);
    layer(x1, 128, 256, W2, b2, g2, be2, x2);
    layer(x2, 256, 512, W3, b3, g3, be3, x3);
}